// R_GCN_Layer_76785425318239
// MI455X (gfx1250) — compile-verified
//
#include <hip/hip_runtime.h>
#include <hip/hip_bf16.h>

typedef __attribute__((ext_vector_type(2))) float v2f;
typedef __attribute__((ext_vector_type(8))) float v8f;

#define N_NODES 50000
#define N_EDGES 1600000
#define REL_NUM 16
#define D_IN 64
#define D_OUT 64
#define T_COLS (REL_NUM * D_OUT)          // 1024
#define N_TILES_ROW (N_NODES / 16)        // 3125
#define N_TILES_COL (T_COLS / 16)         // 64

// ---------------- Phase 0: zero the output accumulator ----------------
__global__ void rgcn_zero(float* __restrict__ out, int n) {
    int i = blockIdx.x * blockDim.x + threadIdx.x;
    if (i < n) out[i] = 0.0f;
}

// ---------------- Phase 1: T[n, r*64+o] = sum_d h[n,d] * W[r,d,o] -----
// One wave computes one 16x16 tile of T via 16x f32 WMMA (K=64).
__global__ __launch_bounds__(256) void rgcn_transform_wmma(
    const float* __restrict__ h,
    const float* __restrict__ W,
    float* __restrict__ T)
{
    const int lane = threadIdx.x & 31;
    const int wave = threadIdx.x >> 5;
    const int tile = blockIdx.x * 8 + wave;            // 200,000 tiles total
    if (tile >= N_TILES_ROW * N_TILES_COL) return;

    const int node_tile = tile / N_TILES_COL;
    const int col_tile  = tile % N_TILES_COL;
    const int row0 = node_tile * 16;                   // node base
    const int r    = col_tile >> 2;                    // relation
    const int o0   = (col_tile & 3) * 16;              // output-col base

    const int m  = lane & 15;                          // row (A) / col (B,C)
    const int hi = lane >> 4;                          // half-wave selector

    // A: lane holds h[row0+m][k + 2*hi + {0,1}]
    const float* hrow = h + (row0 + m) * D_IN;
    // B: lane holds W[r][k + 2*hi + {0,1}][o0 + m]
    const float* wcol = W + r * (D_IN * D_OUT) + o0 + m;

    v8f acc = {};
#pragma unroll
    for (int k = 0; k < D_IN; k += 4) {
        v2f a;
        a.x = hrow[k + 2 * hi + 0];
        a.y = hrow[k + 2 * hi + 1];
        v2f b;
        b.x = wcol[(k + 2 * hi + 0) * D_OUT];
        b.y = wcol[(k + 2 * hi + 1) * D_OUT];
        // (neg_a, A, neg_b, B, c_mod, C, reuse_a, reuse_b)
        acc = __builtin_amdgcn_wmma_f32_16x16x4_f32(
            false, a, false, b, (short)0, acc, false, false);
    }

    // C/D layout: VGPR j -> M = j + 8*hi, N = m
    float* outp = T + (long long)row0 * T_COLS + r * D_OUT + o0 + m;
#pragma unroll
    for (int j = 0; j < 8; ++j) {
        outp[(long long)(j + 8 * hi) * T_COLS] = acc[j];
    }
}

// ---------------- Phase 2: per-edge gather * norm, atomic scatter -----
__global__ void rgcn_edge_scatter(
    const float* __restrict__ T,
    const float* __restrict__ norm,
    const int* __restrict__ src,
    const int* __restrict__ dst,
    const int* __restrict__ rel,
    float* __restrict__ out)
{
    unsigned int idx = blockIdx.x * blockDim.x + threadIdx.x; // < 102.4M
    unsigned int e = idx >> 6;
    int o = idx & 63;
    if (e >= N_EDGES) return;
    int s  = src[e];
    int d  = dst[e];
    int rt = rel[e];
    float v = T[s * T_COLS + rt * D_OUT + o] * norm[e];
    atomicAdd(out + d * D_OUT + o, v);
}

// ---------------- Fallback (small ws): direct per-edge matvec ---------
__global__ void rgcn_edge_direct(
    const float* __restrict__ h,
    const float* __restrict__ W,
    const float* __restrict__ norm,
    const int* __restrict__ src,
    const int* __restrict__ dst,
    const int* __restrict__ rel,
    float* __restrict__ out)
{
    unsigned int idx = blockIdx.x * blockDim.x + threadIdx.x;
    unsigned int e = idx >> 6;
    int o = idx & 63;
    if (e >= N_EDGES) return;
    const float* hs = h + src[e] * D_IN;
    const float* wc = W + rel[e] * (D_IN * D_OUT) + o;
    float acc = 0.0f;
#pragma unroll
    for (int d = 0; d < D_IN; ++d) acc += hs[d] * wc[d * D_OUT];
    atomicAdd(out + dst[e] * D_OUT + o, acc * norm[e]);
}

// ---------------- Phase 3: ReLU ---------------------------------------
__global__ void rgcn_relu(float* __restrict__ out, int n) {
    int i = blockIdx.x * blockDim.x + threadIdx.x;
    if (i < n) out[i] = fmaxf(out[i], 0.0f);
}

extern "C" void kernel_launch(void* const* d_in, const int* in_sizes, int n_in,
                              void* d_out, int out_size, void* d_ws, size_t ws_size,
                              hipStream_t stream) {
    const float* h    = (const float*)d_in[0];   // [N, 64]
    const float* W    = (const float*)d_in[1];   // [16, 64, 64]
    const float* norm = (const float*)d_in[2];   // [E]
    const int*   src  = (const int*)d_in[3];     // [E]
    const int*   dst  = (const int*)d_in[4];     // [E]
    const int*   rel  = (const int*)d_in[5];     // [E]
    float* out = (float*)d_out;                  // [N, 64]

    const int out_elems = N_NODES * D_OUT;       // 3.2M
    rgcn_zero<<<(out_elems + 255) / 256, 256, 0, stream>>>(out, out_elems);

    const size_t t_bytes = (size_t)N_NODES * T_COLS * sizeof(float); // 204.8 MB
    const unsigned int edge_threads = (unsigned int)N_EDGES * 64u;   // 102.4M

    if (ws_size >= t_bytes) {
        float* T = (float*)d_ws;
        const int n_tiles = N_TILES_ROW * N_TILES_COL;               // 200,000
        rgcn_transform_wmma<<<(n_tiles + 7) / 8, 256, 0, stream>>>(h, W, T);
        rgcn_edge_scatter<<<edge_threads / 256, 256, 0, stream>>>(
            T, norm, src, dst, rel, out);
    } else {
        rgcn_edge_direct<<<edge_threads / 256, 256, 0, stream>>>(
            h, W, norm, src, dst, rel, out);
    }

    rgcn_relu<<<(out_elems + 255) / 256, 256, 0, stream>>>(out, out_elems);
}